// RoutingCapsules_9448928051605
// MI455X (gfx1250) — compile-verified
//
#include <hip/hip_runtime.h>
#include <hip/hip_bf16.h>

typedef float v2f __attribute__((ext_vector_type(2)));
typedef float v8f __attribute__((ext_vector_type(8)));

#define B_SZ   512
#define NCAPS  10
#define CCAPS  1152
#define DCAPS  16
#define IDIM   8
#define BT     16            // batch tile = WMMA M
#define MBLK   (B_SZ / BT)   // 32
#define CS     (CCAPS * B_SZ)  // stride of n in [n][c][b] logits/cc layout

// =====================================================================
// s-GEMM:  s[b,n,d] += sum_{c,i} (cc[n,c,b] * x[b,c,i]) * W[n,c,d,i]
// Per workgroup: one batch tile (16 b) x one c-range (48 c), 10 waves (one per n).
// Two independent V_WMMA_F32_16X16X4_F32 chains (M=b, N=d, K=(c,i)).
// UNIFORM template param -> iteration-1 (c = 0.1 uniform) is branch-free.
// =====================================================================
#define S_CR 48
template <int UNIFORM>
__global__ __launch_bounds__(320) void s_gemm_kernel(
    const float* __restrict__ x, const float* __restrict__ W,
    const float* __restrict__ cc, float* __restrict__ s)
{
  __shared__ float x_lds[S_CR * 160];         // [c][b*10 + i]  (stride 10 -> conflict-free b64 reads)
  __shared__ float cc_lds[NCAPS * S_CR * 16]; // [n][c][b]

  const int mb = blockIdx.x;
  const int c0 = blockIdx.y * S_CR;
  const int tid = threadIdx.x;

  // stage x tile: 16b x 48c x 8i = 6144 floats (coalesced global reads)
  for (int t = tid; t < BT * S_CR * IDIM; t += 320) {
    int i = t & 7, cb = t >> 3, c = cb % S_CR, b = cb / S_CR;
    x_lds[c * 160 + b * 10 + i] = x[((mb * BT + b) * CCAPS + (c0 + c)) * IDIM + i];
  }
  // stage c_coef tile: 10n x 48c x 16b = 7680 floats ([n][c][b] source -> 64B chunks)
  if (!UNIFORM) {
    for (int t = tid; t < NCAPS * S_CR * BT; t += 320) {
      int b = t & 15, r = t >> 4, c = r % S_CR, nn = r / S_CR;
      cc_lds[(nn * S_CR + c) * 16 + b] = cc[(nn * CCAPS + c0 + c) * B_SZ + mb * BT + b];
    }
  }
  __syncthreads();

  const int wave = tid >> 5, lane = tid & 31;
  const int n  = wave;                 // wave w handles output capsule n=w
  const int bl = lane & 15;            // A-row b  /  B-col d (per ISA f32 layouts)
  const int kk = (lane >> 4) << 1;     // K pair: lanes 0-15 -> K{0,1}, 16-31 -> K{2,3}
  const float2* __restrict__ W2 = reinterpret_cast<const float2*>(W);

  v8f acc0 = {}, acc1 = {};            // two chains -> 2x WMMA-pipe ILP
#pragma unroll 4
  for (int c = 0; c < S_CR; ++c) {
    float ccv = UNIFORM ? 0.1f : cc_lds[(n * S_CR + c) * 16 + bl];
    const float2* xp = reinterpret_cast<const float2*>(x_lds + c * 160 + bl * 10);
    int wrow = ((n * CCAPS + (c0 + c)) * DCAPS + bl) * IDIM;  // W[n,c,d=bl,0]
    // K-chunk 0: i in 0..3
    {
      float2 xv = xp[kk >> 1];
      v2f a; a.x = ccv * xv.x; a.y = ccv * xv.y;
      float2 wv = W2[(wrow + kk) >> 1];
      v2f bf; bf.x = wv.x; bf.y = wv.y;
      acc0 = __builtin_amdgcn_wmma_f32_16x16x4_f32(false, a, false, bf, (short)0, acc0, false, false);
    }
    // K-chunk 1: i in 4..7
    {
      float2 xv = xp[(4 + kk) >> 1];
      v2f a; a.x = ccv * xv.x; a.y = ccv * xv.y;
      float2 wv = W2[(wrow + 4 + kk) >> 1];
      v2f bf; bf.x = wv.x; bf.y = wv.y;
      acc1 = __builtin_amdgcn_wmma_f32_16x16x4_f32(false, a, false, bf, (short)0, acc1, false, false);
    }
  }
  // C/D layout: lane N=l&15 (=d); VGPR r -> M = r + (l>=16 ? 8 : 0) (=b)
  const int bh = (lane >> 4) * 8, d = lane & 15;
  for (int r = 0; r < 8; ++r) {
    int b = bh + r;
    atomicAdd(&s[((mb * BT + b) * NCAPS + n) * DCAPS + d], acc0[r] + acc1[r]);
  }
}

// =====================================================================
// uv-GEMM (transposed): zT[(c,i), b] = sum_d W[n,c,d,i] * v[b,n,d]
//   WMMA: M=(2c x 8i), N=b, K=d  -> lane holds b=l&15, c~=l>>4, i runs down
//   the 8 accumulator VGPRs, so the i-dot with x is 8 in-lane FMAs (no shuffles).
// logits[n,c,b] += sum_i x[b,c,i] * zT[(c,i),b]   (coalesced RMW, unique owner)
// =====================================================================
#define UV_CB 64   // c per block
#define UV_CW 8    // c per wave (4 z-tiles of 2c)
__global__ __launch_bounds__(256) void uv_kernel(
    const float* __restrict__ x, const float* __restrict__ W,
    const float* __restrict__ vbuf, float* __restrict__ logits)
{
  __shared__ float v_lds[BT * 18];       // [b][d], stride 18 -> conflict-free
  __shared__ float x_lds[UV_CB * 160];   // [c][b*10 + i]

  const int mb = blockIdx.x, n = blockIdx.y, c0 = blockIdx.z * UV_CB;
  const int tid = threadIdx.x;

  {
    int b = tid >> 4, dd = tid & 15;
    v_lds[b * 18 + dd] = vbuf[((mb * BT + b) * NCAPS + n) * DCAPS + dd];
  }
  for (int t = tid; t < BT * UV_CB * IDIM; t += 256) {
    int i = t & 7, cb = t >> 3, c = cb & (UV_CB - 1), b = cb >> 6;
    x_lds[c * 160 + b * 10 + i] = x[((mb * BT + b) * CCAPS + (c0 + c)) * IDIM + i];
  }
  __syncthreads();

  const int wave = tid >> 5, lane = tid & 31;
  const int kk  = (lane >> 4) << 1;     // K pair split across lane halves
  const int mA  = lane & 15;            // A-row (c~,i):  c~ = mA>>3, i = mA&7
  const int ctA = mA >> 3, iA = mA & 7;
  const int bN  = lane & 15;            // B-col / C-col: b
  const int cw  = lane >> 4;            // c~ owned by this lane in C layout

#pragma unroll
  for (int t = 0; t < UV_CW / 2; ++t) {
    int crel = wave * UV_CW + t * 2;    // local c of this 2c tile
    int cg = c0 + crel;
    int arow = ((n * CCAPS + cg + ctA) * DCAPS) * IDIM + iA;  // W[n, c+ctA, 0, iA]
    v8f z0 = {}, z1 = {};               // two chains of 2 WMMAs each
#pragma unroll
    for (int dq = 0; dq < DCAPS; dq += 8) {
      {
        int d0 = dq + kk;
        v2f a; a.x = W[arow + d0 * IDIM]; a.y = W[arow + (d0 + 1) * IDIM];
        const float* vp = v_lds + bN * 18 + d0;
        v2f bf; bf.x = vp[0]; bf.y = vp[1];
        z0 = __builtin_amdgcn_wmma_f32_16x16x4_f32(false, a, false, bf, (short)0, z0, false, false);
      }
      {
        int d0 = dq + 4 + kk;
        v2f a; a.x = W[arow + d0 * IDIM]; a.y = W[arow + (d0 + 1) * IDIM];
        const float* vp = v_lds + bN * 18 + d0;
        v2f bf; bf.x = vp[0]; bf.y = vp[1];
        z1 = __builtin_amdgcn_wmma_f32_16x16x4_f32(false, a, false, bf, (short)0, z1, false, false);
      }
    }
    // lane l: uv[b = l&15, c = cg + (l>>4)] = sum_r (z0[r]+z1[r]) * x[b, c, i=r]
    const float* xrow = x_lds + (crel + cw) * 160 + bN * 10;
    float uv = 0.f;
#pragma unroll
    for (int r = 0; r < 8; ++r) uv += (z0[r] + z1[r]) * xrow[r];
    int li = (n * CCAPS + cg + cw) * B_SZ + mb * BT + bN;  // unique owner, coalesced
    logits[li] += uv;
  }
}

// softmax over n for each (b,c); logits/cc layout [n][c][b] -> fully coalesced
__global__ __launch_bounds__(256) void softmax_kernel(
    const float* __restrict__ logits, float* __restrict__ cc)
{
  int t = blockIdx.x * blockDim.x + threadIdx.x;
  if (t >= B_SZ * CCAPS) return;
  int base = t;  // = c*512 + b
  float v[NCAPS], m = -1e30f;
  for (int nn = 0; nn < NCAPS; ++nn) { v[nn] = logits[nn * CS + base]; m = fmaxf(m, v[nn]); }
  float sum = 0.f;
  for (int nn = 0; nn < NCAPS; ++nn) { v[nn] = __expf(v[nn] - m); sum += v[nn]; }
  float inv = 1.f / sum;
  for (int nn = 0; nn < NCAPS; ++nn) cc[nn * CS + base] = v[nn] * inv;
}

// squash per (b,n); optionally re-zero s so the next atomic s-GEMM starts clean
__global__ __launch_bounds__(256) void squash_kernel(
    float* __restrict__ s, float* __restrict__ dst, int clear)
{
  int t = blockIdx.x * blockDim.x + threadIdx.x;
  if (t >= B_SZ * NCAPS) return;
  int base = t * DCAPS;
  float sv[DCAPS], sq = 0.f;
  const float4* s4 = reinterpret_cast<const float4*>(s + base);
  for (int q = 0; q < 4; ++q) {
    float4 f = s4[q];
    sv[q*4+0] = f.x; sv[q*4+1] = f.y; sv[q*4+2] = f.z; sv[q*4+3] = f.w;
    sq += f.x*f.x + f.y*f.y + f.z*f.z + f.w*f.w;
  }
  float scale = sq / ((1.f + sq) * (sqrtf(sq) + 1e-8f));
  for (int d = 0; d < DCAPS; ++d) dst[base + d] = sv[d] * scale;
  if (clear) for (int d = 0; d < DCAPS; ++d) s[base + d] = 0.f;
}

__global__ __launch_bounds__(256) void zero_kernel(float* __restrict__ p, int nelem)
{
  for (int i = blockIdx.x * blockDim.x + threadIdx.x; i < nelem; i += gridDim.x * blockDim.x)
    p[i] = 0.f;
}

extern "C" void kernel_launch(void* const* d_in, const int* in_sizes, int n_in,
                              void* d_out, int out_size, void* d_ws, size_t ws_size,
                              hipStream_t stream) {
  const float* x = (const float*)d_in[0];   // [512,1152,8]
  const float* W = (const float*)d_in[1];   // [10,1152,16,8]
  float* out = (float*)d_out;               // [512,10,16]

  const int NL = NCAPS * CS;                // 5,898,240  ([n][c][b])
  const int NS = B_SZ * NCAPS * DCAPS;      // 81,920
  float* logits = (float*)d_ws;             // f32, 23.6 MB (persists across iters)
  float* ccoef  = logits + NL;              // f32, 23.6 MB
  float* sbuf   = ccoef + NL;               // f32, 327 KB (atomic accumulator)
  float* vbuf   = sbuf + NS;                // f32, 327 KB

  dim3 sgrid(MBLK, CCAPS / S_CR);           // 32 x 24 wgs, 10 waves each
  dim3 ugrid(MBLK, NCAPS, CCAPS / UV_CB);   // 32 x 10 x 18 wgs, 8 waves each
  int sqb = (B_SZ * NCAPS + 255) / 256;
  int smb = (B_SZ * CCAPS + 255) / 256;

  zero_kernel<<<2048, 256, 0, stream>>>(logits, NL);
  zero_kernel<<<320,  256, 0, stream>>>(sbuf, NS);

  // iter 1: softmax(0) == uniform 0.1
  s_gemm_kernel<1><<<sgrid, 320, 0, stream>>>(x, W, ccoef, sbuf);
  squash_kernel<<<sqb, 256, 0, stream>>>(sbuf, vbuf, 1);
  uv_kernel<<<ugrid, 256, 0, stream>>>(x, W, vbuf, logits);
  softmax_kernel<<<smb, 256, 0, stream>>>(logits, ccoef);
  // iter 2
  s_gemm_kernel<0><<<sgrid, 320, 0, stream>>>(x, W, ccoef, sbuf);
  squash_kernel<<<sqb, 256, 0, stream>>>(sbuf, vbuf, 1);
  uv_kernel<<<ugrid, 256, 0, stream>>>(x, W, vbuf, logits);
  softmax_kernel<<<smb, 256, 0, stream>>>(logits, ccoef);
  // iter 3 (final): squash straight into d_out
  s_gemm_kernel<0><<<sgrid, 320, 0, stream>>>(x, W, ccoef, sbuf);
  squash_kernel<<<sqb, 256, 0, stream>>>(sbuf, out, 0);
}